// GCNEncoder_26792005993094
// MI455X (gfx1250) — compile-verified
//
#include <hip/hip_runtime.h>

typedef __attribute__((ext_vector_type(2))) float v2f;
typedef __attribute__((ext_vector_type(8))) float v8f;

#define LDS_STRIDE 68  // 64 + 4 pad: bank-conflict-free A-frag reads, 16B aligned rows

__global__ void zero_f32(float* __restrict__ p, int n) {
    int i = blockIdx.x * blockDim.x + threadIdx.x;
    if (i < n) p[i] = 0.0f;
}

__global__ void degree_kernel(const int* __restrict__ src, const int* __restrict__ dst,
                              float* __restrict__ outdeg, float* __restrict__ indeg, int E) {
    int e = blockIdx.x * blockDim.x + threadIdx.x;
    if (e < E) {
        __hip_atomic_fetch_add(&outdeg[src[e]], 1.0f, __ATOMIC_RELAXED, __HIP_MEMORY_SCOPE_AGENT);
        __hip_atomic_fetch_add(&indeg[dst[e]], 1.0f, __ATOMIC_RELAXED, __HIP_MEMORY_SCOPE_AGENT);
    }
}

// in-place: deg -> clip(deg,1)^-0.5
__global__ void norm_kernel(float* __restrict__ d, int n) {
    int i = blockIdx.x * blockDim.x + threadIdx.x;
    if (i < n) {
        float v = fmaxf(d[i], 1.0f);
        d[i] = 1.0f / sqrtf(v);
    }
}

// agg[dst] += X[src] * outnorm[src], feature dim = 64, one thread per (edge, feature)
__global__ void spmm_kernel(const float* __restrict__ X, const float* __restrict__ outnorm,
                            const int* __restrict__ src, const int* __restrict__ dst,
                            float* __restrict__ agg, int E) {
    int t = blockIdx.x * blockDim.x + threadIdx.x;
    int e = t >> 6;
    int f = t & 63;
    if (e < E) {
        int s = src[e];
        int d = dst[e];
        float v = X[(size_t)s * 64 + f] * outnorm[s];
        __hip_atomic_fetch_add(&agg[(size_t)d * 64 + f], v,
                               __ATOMIC_RELAXED, __HIP_MEMORY_SCOPE_AGENT);
    }
}

// Y[16 rows] = relu( (X_tile * innorm) @ W + b ), IN=64 fixed, OUT in {32,64}
// one wave32 per 16-row tile, fp32 WMMA 16x16x4
template <int OUT>
__global__ __launch_bounds__(32)
void gemm_bias_relu_kernel(const float* __restrict__ X,      // [n,64]
                           const float* __restrict__ innorm, // [n]
                           const float* __restrict__ W,      // [64,OUT] row-major
                           const float* __restrict__ bias,   // [OUT]
                           float* __restrict__ Y,            // [n,OUT]
                           int n) {
    __shared__ float xs[16 * LDS_STRIDE];
    const int r0   = blockIdx.x * 16;
    const int lane = threadIdx.x;

    // stage 16x64 tile, pre-scaled by innorm
    for (int i = lane; i < 16 * 16; i += 32) {  // 16 rows x 16 float4
        int row  = i >> 4;
        int col4 = (i & 15) << 2;
        int gr   = r0 + row;
        float4 v = make_float4(0.f, 0.f, 0.f, 0.f);
        float  s = 0.f;
        if (gr < n) {
            v = *reinterpret_cast<const float4*>(X + (size_t)gr * 64 + col4);
            s = innorm[gr];
        }
        float* p = &xs[row * LDS_STRIDE + col4];
        p[0] = v.x * s; p[1] = v.y * s; p[2] = v.z * s; p[3] = v.w * s;
    }
    __syncthreads();

    const int half = lane >> 4;   // K sub-pair selector
    const int l16  = lane & 15;

    v8f acc[OUT / 16] = {};

#pragma unroll
    for (int k = 0; k < 64; k += 4) {
        // A 16x4 f32 frag: lanes 0-15 hold row l16 K={k,k+1}; lanes 16-31 K={k+2,k+3}
        const float* xp = &xs[l16 * LDS_STRIDE + k + 2 * half];
        v2f a;
        a.x = xp[0];
        a.y = xp[1];
#pragma unroll
        for (int t = 0; t < OUT / 16; ++t) {
            int col = t * 16 + l16;
            // B 4x16 f32 frag: mirrored layout of A
            v2f b;
            b.x = W[(size_t)(k + 2 * half) * OUT + col];
            b.y = W[(size_t)(k + 2 * half + 1) * OUT + col];
            acc[t] = __builtin_amdgcn_wmma_f32_16x16x4_f32(
                /*neg_a=*/false, a, /*neg_b=*/false, b,
                /*c_mod=*/(short)0, acc[t], /*reuse_a=*/false, /*reuse_b=*/false);
        }
    }

    // C/D layout: vgpr i -> M = i + 8*half, N = l16
#pragma unroll
    for (int t = 0; t < OUT / 16; ++t) {
        int   col = t * 16 + l16;
        float bv  = bias[col];
#pragma unroll
        for (int i = 0; i < 8; ++i) {
            int gr = r0 + half * 8 + i;
            if (gr < n) {
                float v = acc[t][i] + bv;
                Y[(size_t)gr * OUT + col] = v > 0.f ? v : 0.f;
            }
        }
    }
}

extern "C" void kernel_launch(void* const* d_in, const int* in_sizes, int n_in,
                              void* d_out, int out_size, void* d_ws, size_t ws_size,
                              hipStream_t stream) {
    const float* feats = (const float*)d_in[0];
    const int*   src   = (const int*)d_in[1];
    const int*   dst   = (const int*)d_in[2];
    const float* W1    = (const float*)d_in[3];
    const float* b1    = (const float*)d_in[4];
    const float* W2    = (const float*)d_in[5];
    const float* b2    = (const float*)d_in[6];

    const int N = in_sizes[0] / 64;   // 100000
    const int E = in_sizes[1];        // 1600000

    // workspace layout (floats): [outnorm N][innorm N][agg N*64][h N*64]
    float* ws      = (float*)d_ws;
    float* outnorm = ws;
    float* innorm  = ws + N;
    float* agg     = ws + 2 * (size_t)N;
    float* h       = agg + (size_t)N * 64;

    const int aggN  = N * 64;         // 6.4M
    const int spmmT = E * 64;         // 102.4M threads, fits int32

    // degrees -> symmetric norms
    zero_f32<<<(2 * N + 255) / 256, 256, 0, stream>>>(outnorm, 2 * N);
    degree_kernel<<<(E + 255) / 256, 256, 0, stream>>>(src, dst, outnorm, innorm, E);
    norm_kernel<<<(2 * N + 255) / 256, 256, 0, stream>>>(outnorm, 2 * N);

    // layer 1: feats -> h  (64 -> 64)
    zero_f32<<<(aggN + 255) / 256, 256, 0, stream>>>(agg, aggN);
    spmm_kernel<<<(spmmT + 255) / 256, 256, 0, stream>>>(feats, outnorm, src, dst, agg, E);
    gemm_bias_relu_kernel<64><<<(N + 15) / 16, 32, 0, stream>>>(agg, innorm, W1, b1, h, N);

    // layer 2: h -> out  (64 -> 32), reuse agg buffer
    zero_f32<<<(aggN + 255) / 256, 256, 0, stream>>>(agg, aggN);
    spmm_kernel<<<(spmmT + 255) / 256, 256, 0, stream>>>(h, outnorm, src, dst, agg, E);
    gemm_bias_relu_kernel<32><<<(N + 15) / 16, 32, 0, stream>>>(agg, innorm, W2, b2,
                                                                (float*)d_out, N);
}